// TTNModel_1992864825382
// MI455X (gfx1250) — compile-verified
//
#include <hip/hip_runtime.h>

// ---------------------------------------------------------------------------
// TTN forward for MI455X (gfx1250, wave32).
//
// Pipeline per 16-sample batch tile (one 128-thread workgroup, 4 waves):
//   phase1 (VALU): leaf merge  data[B,256,2] -> x1[16b][128n][4]      (LDS)
//   phase2 (WMMA): 4->16 layer (64 nodes, 4 x wmma_f32_16x16x4 each) -> bufA
//   phase3 (WMMA): five 16->16 layers (32/16/8/4/2 nodes,
//                  64 x wmma_f32_16x16x4 each), ping-pong bufA/bufB
//   phase4 (VALU): top contraction -> out[B,2]
//
// Node GEMM mapping:  out[b,k] = sum_pq xl[b,p] xr[b,q] W[pq,k]
//   = [16 x 256] @ [256 x 16] GEMM; A-chunk (K=4) built on the fly with
//   one v_pk_mul_f32 per WMMA; B pre-reordered into WMMA-B lane order in
//   d_ws, packed so one global_load_b128 feeds TWO WMMAs.
// ---------------------------------------------------------------------------

typedef __attribute__((ext_vector_type(2))) float v2f;
typedef __attribute__((ext_vector_type(4))) float v4f;
typedef __attribute__((ext_vector_type(8))) float v8f;

#define N_TILES     4096          // 65536 / 16
#define MID_NODES   62            // 32+16+8+4+2 (16-d layers, merge order L5..L1)
#define WRE_NODE_F  4096          // 32 chunk-pairs * 32 lanes * 4 floats
#define WRE6_BASE_F (MID_NODES * WRE_NODE_F)   // 253952 floats
#define WRE6_NODE_F 256           // 2 chunk-pairs * 32 lanes * 4 floats
// total d_ws usage: 270336 floats = 1,081,344 bytes

// --------------------------- prep: reorder weights -------------------------
// Per chunk c (K=4 slice), lane needs float2 = W[p, qa, k], W[p, qa+1, k]
//   p = c>>2, qa = 4*(c&3) + 2*(lane>>4), k = lane&15     (W is [16][16][16])
// Packed layout: node base + (c>>1)*128 + lane*4 + 2*(c&1)  -> float4 per
// lane covers chunks {2j, 2j+1} so the main kernel loads b128.
__global__ __launch_bounds__(256)
void ttn_prep(const float* __restrict__ w5, const float* __restrict__ w4,
              const float* __restrict__ w3, const float* __restrict__ w2,
              const float* __restrict__ w1, const float* __restrict__ w6,
              float* __restrict__ ws) {
  int t = blockIdx.x * blockDim.x + threadIdx.x;
  const int NM = MID_NODES * 64 * 32;       // 126976 mid lane-tasks
  if (t < NM) {
    int nid  = t >> 11;                     // / (64*32)
    int rem  = t & 2047;
    int c    = rem >> 5;
    int lane = rem & 31;
    int m; const float* src;
    if (nid < 32)      { m = nid;      src = w5; }   // 32 nodes (L5 merge)
    else if (nid < 48) { m = nid - 32; src = w4; }   // 16 nodes
    else if (nid < 56) { m = nid - 48; src = w3; }   // 8
    else if (nid < 60) { m = nid - 56; src = w2; }   // 4
    else               { m = nid - 60; src = w1; }   // 2
    src += (size_t)m * 4096;
    int hi = lane >> 4, k = lane & 15;
    int p  = c >> 2;
    int qa = 4 * (c & 3) + 2 * hi;
    float v0 = src[(p * 16 + qa) * 16 + k];
    float v1 = src[(p * 16 + qa + 1) * 16 + k];
    float* dst = ws + (size_t)nid * WRE_NODE_F
               + (c >> 1) * 128 + lane * 4 + 2 * (c & 1);
    dst[0] = v0; dst[1] = v1;
  } else {
    int u = t - NM;                         // 64*4*32 = 8192 L6 lane-tasks
    if (u < 64 * 4 * 32) {
      int m    = u >> 7;
      int rem  = u & 127;
      int c    = rem >> 5;
      int lane = rem & 31;
      int hi = lane >> 4, k = lane & 15;
      const float* src = w6 + (size_t)m * 256;       // W6[m] is [4][4][16]
      float v0 = src[(c * 4 + 2 * hi) * 16 + k];
      float v1 = src[(c * 4 + 2 * hi + 1) * 16 + k];
      float* dst = ws + WRE6_BASE_F + (size_t)m * WRE6_NODE_F
                 + (c >> 1) * 128 + lane * 4 + 2 * (c & 1);
      dst[0] = v0; dst[1] = v1;
    }
  }
}

// ------------------------------ main fused kernel --------------------------
__global__ __launch_bounds__(128)
void ttn_main(const float* __restrict__ data, const float* __restrict__ leaf_w,
              const float* __restrict__ top_w, const float* __restrict__ wre,
              float* __restrict__ out) {
  // LDS: bufA 64*16*20 f (80KB) | region B 10240 f (40KB) hosting bufB and x1
  __shared__ float smem[20480 + 10240];     // 122,880 bytes
  float* bufA = smem;
  float* bufB = smem + 20480;               // [n][16 rows][20] padded
  float* x1   = bufB;                       // [16 b][516] : b*516 + n*4 + k

  const int tile = blockIdx.x;
  const int b0   = tile << 4;
  const int tid  = threadIdx.x;
  const int wave = tid >> 5;
  const int lane = tid & 31;
  const int row  = lane & 15;               // WMMA M row / batch row
  const int hi   = lane >> 4;               // wave half

  // ---- phase 1: leaf merge (VALU), coalesced 16B global reads -------------
  for (int t = 0; t < 16; ++t) {
    int idx = t * 128 + tid;
    int n = idx & 127;                      // leaf node
    int b = idx >> 7;                       // row in tile
    v4f dv = *(const v4f*)(data + (size_t)(b0 + b) * 512 + n * 4);
    const float* lw = leaf_w + n * 16;      // [p][q][k]
    float p0 = dv.x * dv.z, p1 = dv.x * dv.w;
    float p2 = dv.y * dv.z, p3 = dv.y * dv.w;
    v4f r;
    r.x = p0 * lw[0] + p1 * lw[4] + p2 * lw[8]  + p3 * lw[12];
    r.y = p0 * lw[1] + p1 * lw[5] + p2 * lw[9]  + p3 * lw[13];
    r.z = p0 * lw[2] + p1 * lw[6] + p2 * lw[10] + p3 * lw[14];
    r.w = p0 * lw[3] + p1 * lw[7] + p2 * lw[11] + p3 * lw[15];
    *(v4f*)(x1 + b * 516 + n * 4) = r;
  }
  __syncthreads();

  // ---- phase 2: 4->16 layer, 64 nodes, 4 WMMAs each -> bufA ---------------
  {
    const float* wre6 = wre + WRE6_BASE_F;
    for (int r6 = 0; r6 < 16; ++r6) {
      int m = wave * 16 + r6;
      v4f xl = *(const v4f*)(x1 + row * 516 + (2 * m) * 4);
      v2f xr = *(const v2f*)(x1 + row * 516 + (2 * m + 1) * 4 + 2 * hi);
      const float* wb = wre6 + (size_t)m * WRE6_NODE_F + lane * 4;
      v4f bq0 = *(const v4f*)(wb);          // chunks 0,1
      v4f bq1 = *(const v4f*)(wb + 128);    // chunks 2,3
      v8f acc = {};
      v2f a, bv;
      a.x = xl.x * xr.x; a.y = xl.x * xr.y; bv.x = bq0.x; bv.y = bq0.y;
      acc = __builtin_amdgcn_wmma_f32_16x16x4_f32(false, a, false, bv, (short)0, acc, false, false);
      a.x = xl.y * xr.x; a.y = xl.y * xr.y; bv.x = bq0.z; bv.y = bq0.w;
      acc = __builtin_amdgcn_wmma_f32_16x16x4_f32(false, a, false, bv, (short)0, acc, false, false);
      a.x = xl.z * xr.x; a.y = xl.z * xr.y; bv.x = bq1.x; bv.y = bq1.y;
      acc = __builtin_amdgcn_wmma_f32_16x16x4_f32(false, a, false, bv, (short)0, acc, false, false);
      a.x = xl.w * xr.x; a.y = xl.w * xr.y; bv.x = bq1.z; bv.y = bq1.w;
      acc = __builtin_amdgcn_wmma_f32_16x16x4_f32(false, a, false, bv, (short)0, acc, false, false);
      float* o = bufA + m * 320 + row;      // D: (M = v + 8*hi, N = row)
#pragma unroll
      for (int v = 0; v < 8; ++v) o[(v + 8 * hi) * 20] = acc[v];
    }
  }
  __syncthreads();

  // ---- phase 3: five 16->16 layers, 64 WMMAs per node ---------------------
  const int nOut[5]    = {32, 16, 8, 4, 2};
  const int nidBase[5] = {0, 32, 48, 56, 60};
  for (int li = 0; li < 5; ++li) {
    float* src = (li & 1) ? bufB : bufA;
    float* dst = (li & 1) ? bufA : bufB;
    int n = nOut[li];
    for (int m = wave; m < n; m += 4) {     // wave-uniform; EXEC stays full
      float xls[16];
      {
        const float* xp = src + (2 * m) * 320 + row * 20;
        *(v4f*)(xls + 0)  = *(const v4f*)(xp + 0);
        *(v4f*)(xls + 4)  = *(const v4f*)(xp + 4);
        *(v4f*)(xls + 8)  = *(const v4f*)(xp + 8);
        *(v4f*)(xls + 12) = *(const v4f*)(xp + 12);
      }
      v2f xrp[4];
      {
        const float* xq = src + (2 * m + 1) * 320 + row * 20 + 2 * hi;
        xrp[0] = *(const v2f*)(xq + 0);
        xrp[1] = *(const v2f*)(xq + 4);
        xrp[2] = *(const v2f*)(xq + 8);
        xrp[3] = *(const v2f*)(xq + 12);
      }
      const float* wb = wre + (size_t)(nidBase[li] + m) * WRE_NODE_F + lane * 4;
      v8f acc = {};
#pragma unroll
      for (int j = 0; j < 32; ++j) {        // each j: one b128 -> two WMMAs
        v4f bq = *(const v4f*)(wb + j * 128);
        int c0 = 2 * j, c1 = 2 * j + 1;
        v2f a0, b0, a1, b1;
        a0.x = xls[c0 >> 2] * xrp[c0 & 3].x;
        a0.y = xls[c0 >> 2] * xrp[c0 & 3].y;
        b0.x = bq.x; b0.y = bq.y;
        acc = __builtin_amdgcn_wmma_f32_16x16x4_f32(
            false, a0, false, b0, (short)0, acc, false, false);
        a1.x = xls[c1 >> 2] * xrp[c1 & 3].x;
        a1.y = xls[c1 >> 2] * xrp[c1 & 3].y;
        b1.x = bq.z; b1.y = bq.w;
        acc = __builtin_amdgcn_wmma_f32_16x16x4_f32(
            false, a1, false, b1, (short)0, acc, false, false);
      }
      float* o = dst + m * 320 + row;
#pragma unroll
      for (int v = 0; v < 8; ++v) o[(v + 8 * hi) * 20] = acc[v];
    }
    __syncthreads();
  }

  // ---- phase 4: top contraction (final x in bufB nodes 0,1) ---------------
  if (wave == 0) {
    int b = row, k = hi;                    // 16 rows x 2 labels = 32 lanes
    const float* xl = bufB + b * 20;        // node 0
    const float* xr = bufB + 320 + b * 20;  // node 1
    float acc = 0.f;
#pragma unroll 4
    for (int p = 0; p < 16; ++p) {
      float xlv = xl[p];
      for (int q = 0; q < 16; ++q)
        acc += xlv * xr[q] * top_w[(p * 16 + q) * 2 + k];
    }
    out[(size_t)(b0 + b) * 2 + k] = acc;
  }
}

// ------------------------------- launch ------------------------------------
extern "C" void kernel_launch(void* const* d_in, const int* in_sizes, int n_in,
                              void* d_out, int out_size, void* d_ws, size_t ws_size,
                              hipStream_t stream) {
  (void)in_sizes; (void)n_in; (void)out_size; (void)ws_size;
  const float* data  = (const float*)d_in[0];
  const float* leafw = (const float*)d_in[1];
  const float* w1    = (const float*)d_in[2];   // mid_ws l=1 [2,16,16,16]
  const float* w2    = (const float*)d_in[3];   // l=2 [4,...]
  const float* w3    = (const float*)d_in[4];   // l=3 [8,...]
  const float* w4    = (const float*)d_in[5];   // l=4 [16,...]
  const float* w5    = (const float*)d_in[6];   // l=5 [32,...]
  const float* w6    = (const float*)d_in[7];   // l=6 [64,4,4,16]
  const float* topw  = (const float*)d_in[8];
  float* ws = (float*)d_ws;                     // needs >= 1,081,344 bytes

  // 135,168 lane-tasks = 528 * 256
  ttn_prep<<<528, 256, 0, stream>>>(w5, w4, w3, w2, w1, w6, ws);
  ttn_main<<<N_TILES, 128, 0, stream>>>(data, leafw, topw, ws, (float*)d_out);
}